// ProposalCreator_1683627180553
// MI455X (gfx1250) — compile-verified
//
#include <hip/hip_runtime.h>

#define NPRE   6000
#define NPOST  300
#define WPR    188          // (6000+31)/32 mask words per row
#define BINS   16384
#define CAND   8192
#define CTILE  2048         // column boxes staged per block (32KB LDS)
#define NMS_T  0.7f

typedef __attribute__((ext_vector_type(2))) float v2f;
typedef __attribute__((ext_vector_type(8))) float v8f;
typedef int v4i __attribute__((vector_size(16)));

#if defined(__gfx1250__) && __has_builtin(__builtin_amdgcn_wmma_f32_16x16x4_f32)
#define HAVE_WMMA 1
#else
#define HAVE_WMMA 0
#endif

#if defined(__gfx1250__) && __has_builtin(__builtin_amdgcn_global_load_async_to_lds_b128) && __has_builtin(__builtin_amdgcn_s_wait_asynccnt)
#define HAVE_ASYNC 1
#else
#define HAVE_ASYNC 0
#endif

// ---------------- stage 0: zero scratch that must be clean every call ---------
__global__ void init_kernel(unsigned* hist, unsigned long long* cand) {
    int t = blockIdx.x * blockDim.x + threadIdx.x;
    if (t < BINS) hist[t] = 0u;
    if (t < CAND) cand[t] = 0ull;
}

// ---------------- stage 1: decode boxes, build sortable keys + histogram ------
__global__ void decode_kernel(const float* __restrict__ anchor,
                              const float* __restrict__ loc,
                              const float* __restrict__ score,
                              const int* __restrict__ imgh, const int* __restrict__ imgw,
                              int N, float4* __restrict__ bbox,
                              unsigned* __restrict__ keys, unsigned* __restrict__ hist) {
    int i = blockIdx.x * blockDim.x + threadIdx.x;
    if (i >= N) return;
    float a0 = anchor[i*4+0], a1 = anchor[i*4+1], a2 = anchor[i*4+2], a3 = anchor[i*4+3];
    float srcH = a2 - a0, srcW = a3 - a1;
    float ctrY = a0 + 0.5f*srcH, ctrX = a1 + 0.5f*srcW;
    float dy = loc[i*4+0], dx = loc[i*4+1], dh = loc[i*4+2], dw = loc[i*4+3];
    float cy = dy*srcH + ctrY, cx = dx*srcW + ctrX;
    float h  = expf(dh)*srcH,  w  = expf(dw)*srcW;
    float H = (float)imgh[0], W = (float)imgw[0];
    float y1 = fminf(fmaxf(cy - 0.5f*h, 0.f), H);
    float y2 = fminf(fmaxf(cy + 0.5f*h, 0.f), H);
    float x1 = fminf(fmaxf(cx - 0.5f*w, 0.f), W);
    float x2 = fminf(fmaxf(cx + 0.5f*w, 0.f), W);
    bbox[i] = make_float4(y1, x1, y2, x2);
    bool valid = ((y2 - y1) >= 16.f) && ((x2 - x1) >= 16.f);
    float s = score[i];
    unsigned bits = __float_as_uint(s);
    unsigned key = valid ? ((bits & 0x80000000u) ? ~bits : (bits | 0x80000000u))
                         : 0x007FFFFFu; // key(-inf)
    keys[i] = key;
    atomicAdd(&hist[key >> 18], 1u);
}

// ---------------- stage 2: find threshold bin so that count(bin>=T) >= NPRE ---
__global__ void thresh_kernel(const unsigned* __restrict__ hist, int* meta) {
    __shared__ unsigned csum[256];
    int tid = threadIdx.x;
    int hi = BINS - 1 - 64 * tid;
    unsigned s = 0;
    for (int k = 0; k < 64; ++k) s += hist[hi - k];
    csum[tid] = s;
    __syncthreads();
    if (tid == 0) {
        unsigned cum = 0; int T = 0; bool found = false;
        for (int c = 0; c < 256 && !found; ++c) {
            if (cum + csum[c] >= (unsigned)NPRE) {
                int h2 = BINS - 1 - 64 * c;
                unsigned cum2 = cum;
                for (int k = 0; k < 64; ++k) {
                    cum2 += hist[h2 - k];
                    if (cum2 >= (unsigned)NPRE) { T = h2 - k; break; }
                }
                found = true;
            } else cum += csum[c];
        }
        meta[1] = found ? T : 0;
        meta[0] = 0;   // candidate counter for compact stage
    }
}

// ---------------- stage 3: compact candidates (key | ~index) ------------------
__global__ void compact_kernel(const unsigned* __restrict__ keys, int N,
                               int* meta, unsigned long long* __restrict__ cand) {
    int i = blockIdx.x * blockDim.x + threadIdx.x;
    if (i >= N) return;
    unsigned k = keys[i];
    if ((int)(k >> 18) >= meta[1]) {
        int pos = atomicAdd(&meta[0], 1);
        if (pos < CAND)
            cand[pos] = ((unsigned long long)k << 32) |
                        (unsigned long long)(0xFFFFFFFFu - (unsigned)i);
    }
}

// ---------------- stage 4: single-WGP bitonic sort (desc) of 8192 u64 in LDS --
__global__ void sort_kernel(const unsigned long long* __restrict__ cand,
                            unsigned* __restrict__ topIdx, int N) {
    __shared__ unsigned long long s[CAND];   // 64KB of the 320KB/WGP LDS
    int tid = threadIdx.x;                   // 1024 threads
    for (int i = tid; i < CAND; i += 1024) s[i] = cand[i];
    for (unsigned k = 2; k <= CAND; k <<= 1) {
        for (unsigned jj = k >> 1; jj > 0; jj >>= 1) {
            __syncthreads();
            for (unsigned i = tid; i < CAND; i += 1024) {
                unsigned ixj = i ^ jj;
                if (ixj > i) {
                    bool up = ((i & k) == 0);
                    unsigned long long a = s[i], b = s[ixj];
                    bool sw = up ? (a < b) : (a > b);   // descending sort
                    if (sw) { s[i] = b; s[ixj] = a; }
                }
            }
        }
    }
    __syncthreads();
    for (int j = tid; j < NPRE; j += 1024) {
        unsigned long long v = s[j];
        unsigned key = (unsigned)(v >> 32);
        unsigned idx = 0xFFFFFFFFu - (unsigned)(v & 0xFFFFFFFFull);
        topIdx[j] = (key >= 0x00800000u && idx < (unsigned)N) ? idx : 0xFFFFFFFFu;
    }
}

// ---------------- stage 5: gather top boxes + validity ------------------------
__global__ void gather_kernel(const unsigned* __restrict__ topIdx,
                              const float4* __restrict__ bbox,
                              float4* __restrict__ topBoxes,
                              unsigned* __restrict__ topValid) {
    int j = blockIdx.x * blockDim.x + threadIdx.x;
    if (j >= NPRE) return;
    unsigned id = topIdx[j];
    if (id != 0xFFFFFFFFu) { topBoxes[j] = bbox[id]; topValid[j] = 1u; }
    else { topBoxes[j] = make_float4(0.f,0.f,0.f,0.f); topValid[j] = 0u; }
}

// ---------------- stage 6: IoU suppression-mask matrix ------------------------
// Block: 256 thr = 8 waves; rows/block = 128 (16 rows/wave), cols/block = 2048.
// Per 16x16 tile: denominator area_i + area_j computed by V_WMMA_F32_16X16X4_F32
// (A rows [a_i,1,0,0], B cols [1,a_j,0,0]^T); predicate -> wave32 __ballot;
// two half-tiles recombine into row-major 32-bit mask words.
__global__ void mask_kernel(const float4* __restrict__ boxes,
                            unsigned* __restrict__ mask) {
    __shared__ float4 sbox[CTILE];     // 32KB column tile
    __shared__ float4 srow[8][16];     // 16 row boxes per wave
    const int tid  = threadIdx.x;
    const int wave = tid >> 5, lane = tid & 31;
    const int jTile = blockIdx.y;
    const int jBase = jTile * CTILE;
    const int nj = min(CTILE, NPRE - jBase);

#if HAVE_ASYNC
    for (int t = tid; t < nj; t += 256)
        __builtin_amdgcn_global_load_async_to_lds_b128(
            (__attribute__((address_space(1))) v4i*)(boxes + jBase + t),
            (__attribute__((address_space(3))) v4i*)(&sbox[t]), 0, 0);
    __builtin_amdgcn_s_wait_asynccnt(0);
#else
    for (int t = tid; t < nj; t += 256) sbox[t] = boxes[jBase + t];
#endif

    const int rowBase = blockIdx.x * 128 + wave * 16;
    float areaRow;
    {
        int m = lane & 15;
        int rr = rowBase + m;
        float4 b = (rr < NPRE) ? boxes[rr] : make_float4(0.f,0.f,0.f,0.f);
        if (lane < 16) srow[wave][m] = b;
        areaRow = (b.z - b.x) * (b.w - b.y);
    }
    __syncthreads();

#if HAVE_WMMA
    v2f Aop;
    Aop.x = (lane < 16) ? areaRow : 0.f;   // K=0 slot: area_i
    Aop.y = (lane < 16) ? 1.f     : 0.f;   // K=1 slot: 1
#endif
    const int half = lane >> 4;
    const int nsel = lane & 15;
    const int wlim = (nj + 31) >> 5;

    for (int w = 0; w < wlim; ++w) {
        int c0 = w * 32;
        float4 bj0 = sbox[c0 + nsel];
        float4 bj1 = sbox[c0 + 16 + nsel];
        float aj0 = (bj0.z - bj0.x) * (bj0.w - bj0.y);
        float aj1 = (bj1.z - bj1.x) * (bj1.w - bj1.y);
#if HAVE_WMMA
        v2f B0, B1;
        B0.x = (lane < 16) ? 1.f : 0.f;  B0.y = (lane < 16) ? aj0 : 0.f;
        B1.x = (lane < 16) ? 1.f : 0.f;  B1.y = (lane < 16) ? aj1 : 0.f;
        v8f zero = {};
        v8f cA = __builtin_amdgcn_wmma_f32_16x16x4_f32(false, Aop, false, B0, (short)0, zero, false, false);
        v8f cB = __builtin_amdgcn_wmma_f32_16x16x4_f32(false, Aop, false, B1, (short)0, zero, false, false);
#endif
#pragma unroll
        for (int r = 0; r < 8; ++r) {
            int i = rowBase + r + 8 * half;
            float4 bi = srow[wave][r + 8 * half];
#if HAVE_WMMA
            float sum0 = cA[r], sum1 = cB[r];           // area_i + area_j
#else
            float ai = (bi.z - bi.x) * (bi.w - bi.y);
            float sum0 = ai + aj0, sum1 = ai + aj1;
#endif
            int j0 = jBase + c0 + nsel;
            int j1 = j0 + 16;
            float t0y = fmaxf(bi.x, bj0.x), t0x = fmaxf(bi.y, bj0.y);
            float b0y = fminf(bi.z, bj0.z), b0x = fminf(bi.w, bj0.w);
            float in0 = fmaxf(b0y - t0y, 0.f) * fmaxf(b0x - t0x, 0.f);
            float t1y = fmaxf(bi.x, bj1.x), t1x = fmaxf(bi.y, bj1.y);
            float b1y = fminf(bi.z, bj1.z), b1x = fminf(bi.w, bj1.w);
            float in1 = fmaxf(b1y - t1y, 0.f) * fmaxf(b1x - t1x, 0.f);
            bool p0 = (in0 / (sum0 - in0 + 1e-9f) > NMS_T) && (j0 > i) && (j0 < NPRE);
            bool p1 = (in1 / (sum1 - in1 + 1e-9f) > NMS_T) && (j1 > i) && (j1 < NPRE);
            unsigned bal0 = (unsigned)__ballot(p0);
            unsigned bal1 = (unsigned)__ballot(p1);
            if (lane == 0) {
                unsigned gw = (unsigned)(jTile * (CTILE/32) + w);
                unsigned w0 = (bal0 & 0xFFFFu) | ((bal1 & 0xFFFFu) << 16); // row r
                unsigned w1 = (bal0 >> 16)     | ((bal1 >> 16) << 16);     // row r+8
                int row0 = rowBase + r, row1 = rowBase + r + 8;
                if (row0 < NPRE) mask[row0 * WPR + gw] = w0;
                if (row1 < NPRE) mask[row1 * WPR + gw] = w1;
            }
        }
    }
}

// ---------------- stage 7: serial-chunk suppression scan + emit top 300 -------
__global__ void scan_out_kernel(const unsigned* __restrict__ mask,
                                const unsigned* __restrict__ topValid,
                                const float4* __restrict__ topBoxes,
                                float* __restrict__ out) {
    __shared__ unsigned supp[WPR];
    __shared__ unsigned wordBase[WPR];
    __shared__ unsigned aliveSh;
    int tid = threadIdx.x; // 256
    for (int t = tid; t < NPOST * 4; t += blockDim.x) out[t] = 0.f;
    for (int w = tid; w < WPR; w += blockDim.x) {
        unsigned s = 0;
        for (int k = 0; k < 32; ++k) {
            int j = w * 32 + k;
            unsigned v = (j < NPRE) ? topValid[j] : 0u;
            if (!v) s |= (1u << k);
        }
        supp[w] = s;
    }
    __syncthreads();
    for (int c = 0; c < WPR; ++c) {
        if (tid == 0) {
            unsigned sw = supp[c], alive = 0;
            for (int k = 0; k < 32; ++k) {
                if (!((sw >> k) & 1u)) {
                    alive |= 1u << k;
                    sw |= mask[(c * 32 + k) * WPR + c]; // intra-chunk suppression
                }
            }
            supp[c] = sw;
            aliveSh = alive;
        }
        __syncthreads();
        unsigned alive = aliveSh;
        if (alive) {
            for (int w = c + 1 + tid; w < WPR; w += blockDim.x) {
                unsigned acc = supp[w], a = alive;
                while (a) {
                    int k = __ffs(a) - 1; a &= a - 1;
                    acc |= mask[(c * 32 + k) * WPR + w];
                }
                supp[w] = acc;
            }
        }
        __syncthreads();
    }
    if (tid == 0) {
        unsigned acc = 0;
        for (int w = 0; w < WPR; ++w) {
            wordBase[w] = acc;
            unsigned keepw = ~supp[w];
            if (w == WPR - 1) keepw &= 0xFFFFu;
            acc += __popc(keepw);
        }
    }
    __syncthreads();
    for (int i = tid; i < NPRE; i += blockDim.x) {
        int w = i >> 5, b = i & 31;
        unsigned keepw = ~supp[w];
        if (w == WPR - 1) keepw &= 0xFFFFu;
        if ((keepw >> b) & 1u) {
            unsigned rank = wordBase[w] + __popc(keepw & ((1u << b) - 1u));
            if (rank < NPOST) {
                float4 bx = topBoxes[i];
                out[rank*4+0] = bx.x; out[rank*4+1] = bx.y;
                out[rank*4+2] = bx.z; out[rank*4+3] = bx.w;
            }
        }
    }
}

extern "C" void kernel_launch(void* const* d_in, const int* in_sizes, int n_in,
                              void* d_out, int out_size, void* d_ws, size_t ws_size,
                              hipStream_t stream) {
    const float* anchor = (const float*)d_in[0];
    const float* loc    = (const float*)d_in[1];
    const float* score  = (const float*)d_in[2];
    const int*   imgh   = (const int*)d_in[3];
    const int*   imgw   = (const int*)d_in[4];
    const int N = in_sizes[2];

    char* w = (char*)d_ws; size_t off = 0;
    auto take = [&](size_t bytes) -> void* {
        void* p = w + off; off += (bytes + 255) & ~(size_t)255; return p;
    };
    float4*             bbox     = (float4*)take((size_t)N * 16);
    unsigned*           keys     = (unsigned*)take((size_t)N * 4);
    unsigned*           hist     = (unsigned*)take((size_t)BINS * 4);
    int*                meta     = (int*)take(256);
    unsigned long long* cand     = (unsigned long long*)take((size_t)CAND * 8);
    unsigned*           topIdx   = (unsigned*)take((size_t)NPRE * 4);
    float4*             topBoxes = (float4*)take((size_t)NPRE * 16);
    unsigned*           topValid = (unsigned*)take((size_t)NPRE * 4);
    unsigned*           maskMat  = (unsigned*)take((size_t)NPRE * WPR * 4);
    (void)ws_size; (void)n_in; (void)out_size;

    int nb = (N + 255) / 256;
    init_kernel<<<BINS / 256, 256, 0, stream>>>(hist, cand);
    decode_kernel<<<nb, 256, 0, stream>>>(anchor, loc, score, imgh, imgw, N, bbox, keys, hist);
    thresh_kernel<<<1, 256, 0, stream>>>(hist, meta);
    compact_kernel<<<nb, 256, 0, stream>>>(keys, N, meta, cand);
    sort_kernel<<<1, 1024, 0, stream>>>(cand, topIdx, N);
    gather_kernel<<<(NPRE + 255) / 256, 256, 0, stream>>>(topIdx, bbox, topBoxes, topValid);
    mask_kernel<<<dim3((NPRE + 127) / 128, (NPRE + CTILE - 1) / CTILE), 256, 0, stream>>>(topBoxes, maskMat);
    scan_out_kernel<<<1, 256, 0, stream>>>(maskMat, topValid, topBoxes, (float*)d_out);
}